// CenterLoss_21277267984788
// MI455X (gfx1250) — compile-verified
//
#include <hip/hip_runtime.h>
#include <math.h>

#define CLS_NUM 1000
#define FEAT    512

typedef __attribute__((ext_vector_type(2))) float v2f;
typedef __attribute__((ext_vector_type(8))) float v8f;

// ---------------------------------------------------------------------------
// Kernel 1: zero the class-count buffer (in d_ws) and the output scalar.
// Must run every kernel_launch (graph replays do not re-poison buffers).
// ---------------------------------------------------------------------------
__global__ void centerloss_init_kernel(unsigned int* __restrict__ count,
                                       float* __restrict__ out) {
    int i = blockIdx.x * blockDim.x + threadIdx.x;
    if (i < CLS_NUM) count[i] = 0u;
    if (i == 0) out[0] = 0.0f;
}

// ---------------------------------------------------------------------------
// Kernel 2: histogram of labels. LDS-private histogram per block, then flush
// with global atomics (1000 bins * 4B = 4000B LDS; trivial vs 320KB/WGP).
// ---------------------------------------------------------------------------
__global__ __launch_bounds__(256)
void centerloss_hist_kernel(const int* __restrict__ ys,
                            unsigned int* __restrict__ count, int n) {
    __shared__ unsigned int lcount[CLS_NUM];
    for (int i = threadIdx.x; i < CLS_NUM; i += blockDim.x) lcount[i] = 0u;
    __syncthreads();
    for (int i = blockIdx.x * blockDim.x + threadIdx.x; i < n;
         i += gridDim.x * blockDim.x) {
        atomicAdd(&lcount[ys[i]], 1u);
    }
    __syncthreads();
    for (int i = threadIdx.x; i < CLS_NUM; i += blockDim.x) {
        unsigned int c = lcount[i];
        if (c) atomicAdd(&count[i], c);
    }
}

// ---------------------------------------------------------------------------
// Kernel 3: per-row squared distance via V_WMMA_F32_16X16X4_F32.
// One wave32 owns a tile of 16 rows. A-matrix (16x4 f32): lanes 0-15 hold
// K=k0,k0+1 of row M=lane; lanes 16-31 hold K=k0+2,k0+3 of row M=lane-16.
// B = all-ones -> D[m,n] = sum_k A[m,k] (row sums of squared diffs).
// D layout: lanes 0-15 c[r] = row r; lanes 16-31 c[r] = row 8+r.
// ---------------------------------------------------------------------------
__global__ __launch_bounds__(256)
void centerloss_dist_kernel(const float* __restrict__ xs,
                            const int* __restrict__ ys,
                            const float* __restrict__ center,
                            const unsigned int* __restrict__ count,
                            float* __restrict__ out) {
    const int lane = threadIdx.x & 31;
    const int wave = threadIdx.x >> 5;
    const int tile = blockIdx.x * (blockDim.x >> 5) + wave;   // 16 rows/tile

    const int m      = lane & 15;            // row within tile
    const int kphase = (lane >> 4) << 1;     // 0 for lanes 0-15, 2 for 16-31
    const int row    = tile * 16 + m;

    const int cls = ys[row];
    const float* __restrict__ xrow = xs     + (size_t)row * FEAT + kphase;
    const float* __restrict__ crow = center + (size_t)cls * FEAT + kphase;

    v8f acc0 = {};
    v8f acc1 = {};
    v2f one  = {1.0f, 1.0f};                 // B matrix = all ones

    for (int k = 0; k < FEAT; k += 8) {
        // keep the HBM stream ahead (emits global_prefetch_b8; speculative,
        // OOB-safe per ISA)
        __builtin_prefetch(xrow + k + 256, 0, 0);

        v2f x0 = *(const v2f*)(xrow + k);
        v2f c0 = *(const v2f*)(crow + k);
        v2f x1 = *(const v2f*)(xrow + k + 4);
        v2f c1 = *(const v2f*)(crow + k + 4);

        v2f d0 = x0 - c0;
        v2f d1 = x1 - c1;
        v2f a0 = d0 * d0;
        v2f a1 = d1 * d1;

        // D = A x B + C : row-sum reduction in fp32 on the matrix unit
        acc0 = __builtin_amdgcn_wmma_f32_16x16x4_f32(
            /*neg_a=*/false, a0, /*neg_b=*/false, one,
            /*c_mod=*/(short)0, acc0, /*reuse_a=*/false, /*reuse_b=*/false);
        acc1 = __builtin_amdgcn_wmma_f32_16x16x4_f32(
            false, a1, false, one, (short)0, acc1, false, false);
    }
    v8f acc = acc0 + acc1;

    // lanes 0-15: rows tile*16+0..7 ; lanes 16-31: rows tile*16+8..15
    const int rbase = tile * 16 + ((lane >> 4) << 3);
    float partial = 0.0f;
#pragma unroll
    for (int r = 0; r < 8; ++r) {
        float cnt = (float)count[ys[rbase + r]];   // >= 1 for present labels
        partial += __builtin_sqrtf(acc[r]) / cnt;
    }
    // every lane in each half-group computed the identical partial; commit one
    if ((lane & 15) == 0) atomicAdd(out, partial);
}

// ---------------------------------------------------------------------------
// Host-side launcher
// ---------------------------------------------------------------------------
extern "C" void kernel_launch(void* const* d_in, const int* in_sizes, int n_in,
                              void* d_out, int out_size, void* d_ws, size_t ws_size,
                              hipStream_t stream) {
    const float* xs     = (const float*)d_in[0];
    const int*   ys     = (const int*)d_in[1];
    const float* center = (const float*)d_in[2];
    float*       out    = (float*)d_out;
    unsigned int* count = (unsigned int*)d_ws;   // CLS_NUM * 4 bytes scratch

    const int n = in_sizes[1];                   // N samples

    centerloss_init_kernel<<<(CLS_NUM + 255) / 256, 256, 0, stream>>>(count, out);
    centerloss_hist_kernel<<<64, 256, 0, stream>>>(ys, count, n);

    const int tiles  = n / 16;                   // 16 rows per wave-tile
    const int waves  = 8;                        // 256 threads = 8 wave32
    const int blocks = (tiles + waves - 1) / waves;
    centerloss_dist_kernel<<<blocks, 256, 0, stream>>>(xs, ys, center, count, out);
}